// Decoder_11879879543439
// MI455X (gfx1250) — compile-verified
//
#include <hip/hip_runtime.h>
#include <hip/hip_bf16.h>

// ---------------------------------------------------------------------------
// MI455X (gfx1250) attention-LSTM decoder.
//
// Structure:
//  * logits projection (31.5 GFLOP, 246MB HBM writes -> ~10.5us floor) is
//    hoisted out of the time loop into ONE f32-WMMA GEMM [1920,256]@[256,32000]
//    with W_fc L2-resident. f32 WMMA keeps full precision; still mem-bound.
//  * the 30-step recurrence is latency/L2-stream bound. LSTM gates are
//    column-local (c/h at u need z cols {u,u+256,u+512,u+768}), so the z-GEMM
//    is partitioned 4(batch) x 4(u-group) = 16 WGs/step: same total weight
//    traffic, 4x more WGPs draining it, gates finished WG-locally.
//  * attention step also runs 4x4 WGs; the concat(h,ctx)@W_attn projection
//    uses f32 WMMA from padded LDS instead of redundant VALU dot products.
// ---------------------------------------------------------------------------

constexpr int Bsz  = 64;
constexpr int Tdec = 30;
constexpr int Senc = 31;
constexpr int Edim = 128;
constexpr int Udim = 256;
constexpr int Vdim = 32000;
constexpr int U4   = 4 * Udim;        // 1024
constexpr int KZ   = Edim + 2 * Udim; // 640 = E + U(attn) + U(h)

typedef float v2f __attribute__((ext_vector_type(2)));
typedef float v8f __attribute__((ext_vector_type(8)));

__device__ __forceinline__ v8f wmma_f32_4(v2f a, v2f b, v8f c) {
    // V_WMMA_F32_16X16X4_F32 : D(16x16,f32) = A(16x4,f32) x B(4x16,f32) + C
    return __builtin_amdgcn_wmma_f32_16x16x4_f32(
        /*neg_a=*/false, a, /*neg_b=*/false, b,
        /*c_mod=*/(short)0, c, /*reuse_a=*/false, /*reuse_b=*/false);
}

__device__ __forceinline__ float sigm(float x) {
    return 1.0f / (1.0f + __expf(-x));
}

// --------------------------- init: h, c, attn[0]=0 --------------------------
__global__ void init_kernel(const float* __restrict__ sh, const float* __restrict__ sc,
                            float* __restrict__ h, float* __restrict__ c,
                            float* __restrict__ attn0) {
    int i = blockIdx.x * 256 + threadIdx.x;
    if (i < Bsz * Udim) { h[i] = sh[i]; c[i] = sc[i]; attn0[i] = 0.0f; }
}

// --------------------- keys = memory @ W_mem  [1984,256] ---------------------
// 4 waves/WG, each wave: 16(M) x 64(N) tile, K=256 in steps of 4 (f32 WMMA).
__global__ __launch_bounds__(128) void keys_kernel(const float* __restrict__ mem,
                                                   const float* __restrict__ Wm,
                                                   float* __restrict__ keys) {
    int wave = threadIdx.x >> 5, lane = threadIdx.x & 31;
    int laneM = lane & 15, kHi = lane >> 4;
    int m0 = blockIdx.x * 16;
    int n0 = wave * 64;
    v8f acc[4] = {};
    const float* arow = mem + (m0 + laneM) * Udim;
    for (int k0 = 0; k0 < Udim; k0 += 4) {
        int kk = k0 + 2 * kHi;
        v2f a = *(const v2f*)(arow + kk);
        const float* w0 = Wm + kk * Udim;
        const float* w1 = w0 + Udim;
#pragma unroll
        for (int j = 0; j < 4; ++j) {
            int n = n0 + j * 16 + laneM;
            v2f b; b.x = w0[n]; b.y = w1[n];
            acc[j] = wmma_f32_4(a, b, acc[j]);
        }
    }
#pragma unroll
    for (int j = 0; j < 4; ++j)
#pragma unroll
        for (int r = 0; r < 8; ++r)
            keys[(m0 + r + 8 * kHi) * Udim + n0 + j * 16 + laneM] = acc[j][r];
}

// ----------------------------- LSTM cell step -------------------------------
// Grid (4 batch-groups x 4 u-groups), 256 thr (8 waves). WG (bg,ug) computes
// z for 16 batch rows and the 4 gate-quadrant column slices
// {g*256 + ug*64 + [0,64)}, then finishes its gates locally (column-local).
__global__ __launch_bounds__(256) void lstm_step_kernel(
    const int* __restrict__ inputs, const float* __restrict__ emb,
    const float* __restrict__ Wk, const float* __restrict__ Wr,
    const float* __restrict__ bl, const float* __restrict__ attn_prev,
    float* __restrict__ h, float* __restrict__ c, int t)
{
    const int CSTR = 644;                 // pad: 16 A-rows hit distinct banks
    __shared__ float cin[16 * CSTR];      // 41216 B
    __shared__ float zq[16 * 256];        // [row][gate][64]  16384 B
    int tid = threadIdx.x;
    int bg = blockIdx.x, ug = blockIdx.y;
    int b0 = bg * 16;

    // Stage cell_in = [emb(inputs[b,t]) | attn_prev[b] | h[b]]  (16 x 640)
    for (int idx = tid; idx < 16 * KZ; idx += 256) {
        int r = idx / KZ, k = idx % KZ;
        int b = b0 + r;
        float v;
        if (k < Edim) {
            int tok = inputs[b * Tdec + t];
            v = emb[tok * Edim + k];
        } else if (k < Edim + Udim) {
            v = attn_prev[b * Udim + (k - Edim)];
        } else {
            v = h[b * Udim + (k - Edim - Udim)];
        }
        cin[r * CSTR + k] = v;
    }
    __syncthreads();

    int wave = tid >> 5, lane = tid & 31;
    int laneM = lane & 15, kHi = lane >> 4;
    int q    = wave >> 1;                 // gate quadrant 0..3
    int half = wave & 1;                  // 32-col half within this WG's slice
    int nbase = q * 256 + ug * 64 + half * 32; // global z column base
    v8f acc[2] = {};
    for (int k0 = 0; k0 < KZ; k0 += 4) {
        int kk = k0 + 2 * kHi;
        v2f a = *(const v2f*)(&cin[laneM * CSTR + kk]);
        // virtual weight: rows [0,384) from W_k, rows [384,640) from W_r
        const float* w0 = (k0 < 384) ? (Wk + kk * U4) : (Wr + (kk - 384) * U4);
        const float* w1 = w0 + U4;
#pragma unroll
        for (int j = 0; j < 2; ++j) {
            int n = nbase + j * 16 + laneM;
            v2f b; b.x = w0[n]; b.y = w1[n];
            acc[j] = wmma_f32_4(a, b, acc[j]);
        }
    }
#pragma unroll
    for (int j = 0; j < 2; ++j)
#pragma unroll
        for (int r = 0; r < 8; ++r)
            zq[(r + 8 * kHi) * 256 + q * 64 + half * 32 + j * 16 + laneM] = acc[j][r];
    __syncthreads();

    // gates for u in [ug*64, ug*64+64): c'=s(f)c+s(i)tanh(g); h'=s(o)tanh(c')
    for (int idx = tid; idx < 16 * 64; idx += 256) {
        int r = idx >> 6, j = idx & 63;
        int u = ug * 64 + j;
        int b = b0 + r;
        float zi = zq[r * 256 + 0 * 64 + j] + bl[u];
        float zf = zq[r * 256 + 1 * 64 + j] + bl[u + Udim];
        float zg = zq[r * 256 + 2 * 64 + j] + bl[u + 2 * Udim];
        float zo = zq[r * 256 + 3 * 64 + j] + bl[u + 3 * Udim];
        float cn = sigm(zf) * c[b * Udim + u] + sigm(zi) * tanhf(zg);
        float hn = sigm(zo) * tanhf(cn);
        c[b * Udim + u] = cn;
        h[b * Udim + u] = hn;
    }
}

// ------------------------ attention + attn projection -----------------------
// Grid (4 batch-groups x 4 n-groups), 256 thr. score/softmax/context are tiny
// (duplicated per n-group); the projection concat(h,ctx)@W_attn runs as f32
// WMMA, waves 0..3 each owning one 16-col tile of this WG's 64 output cols.
__global__ __launch_bounds__(256) void attn_step_kernel(
    const float* __restrict__ mem, const float* __restrict__ keys,
    const float* __restrict__ Wa, const float* __restrict__ h,
    float* __restrict__ attn_out)
{
    const int HSTR = 260;                 // pad for bank-conflict-free A frags
    __shared__ float sh_h[16 * HSTR];
    __shared__ float ctx[16 * HSTR];
    __shared__ float sc[16][32];
    int tid = threadIdx.x;
    int bg = blockIdx.x, ng = blockIdx.y;
    int b0 = bg * 16;

    for (int idx = tid; idx < 16 * Udim; idx += 256)
        sh_h[(idx / Udim) * HSTR + (idx % Udim)] =
            h[(b0 + idx / Udim) * Udim + idx % Udim];
    __syncthreads();

    // score[b,s] = h[b] . keys[b,s]
    for (int p = tid; p < 16 * Senc; p += 256) {
        int r = p / Senc, s = p % Senc;
        const float* krow = keys + ((b0 + r) * Senc + s) * Udim;
        float a = 0.0f;
        for (int u = 0; u < Udim; ++u) a += sh_h[r * HSTR + u] * krow[u];
        sc[r][s] = a;
    }
    __syncthreads();
    if (tid < 16) { // softmax over S=31, one thread per row
        float mx = -1e30f;
        for (int s = 0; s < Senc; ++s) mx = fmaxf(mx, sc[tid][s]);
        float sum = 0.0f;
        for (int s = 0; s < Senc; ++s) { float e = __expf(sc[tid][s] - mx); sc[tid][s] = e; sum += e; }
        float inv = 1.0f / sum;
        for (int s = 0; s < Senc; ++s) sc[tid][s] *= inv;
    }
    __syncthreads();
    // context[b] = sum_s align[b,s] * memory[b,s]
    for (int idx = tid; idx < 16 * Udim; idx += 256) {
        int r = idx / Udim, u = idx % Udim;
        const float* mrow = mem + (b0 + r) * Senc * Udim + u;
        float a = 0.0f;
        for (int s = 0; s < Senc; ++s) a += sc[r][s] * mrow[s * Udim];
        ctx[r * HSTR + u] = a;
    }
    __syncthreads();

    // attn = [h | ctx] @ W_attn (K=512) via f32 WMMA; waves 0..3: one 16-col
    // tile each at n = ng*64 + wave*16.
    int wave = tid >> 5, lane = tid & 31;
    int laneM = lane & 15, kHi = lane >> 4;
    if (wave < 4) {
        int n = ng * 64 + wave * 16 + laneM;
        v8f acc = {};
        for (int k0 = 0; k0 < 2 * Udim; k0 += 4) {
            int kk = k0 + 2 * kHi;
            const float* abase = (k0 < Udim) ? (sh_h + (kk))
                                             : (ctx + (kk - Udim));
            v2f a = *(const v2f*)(abase + laneM * HSTR);
            const float* w0 = Wa + kk * Udim;
            const float* w1 = w0 + Udim;
            v2f b; b.x = w0[n]; b.y = w1[n];
            acc = wmma_f32_4(a, b, acc);
        }
#pragma unroll
        for (int r = 0; r < 8; ++r)
            attn_out[(b0 + r + 8 * kHi) * Udim + n] = acc[r];
    }
}

// ----------------- logits: [1920,256] @ [256,32000] + b_fc ------------------
// Row m = t*64 + b.  Wave tile 64x64 (16 v8f accumulators, 4x reuse of both
// A- and B-frags per K-step). 4 waves/WG -> WG covers 64(M) x 256(N).
__global__ __launch_bounds__(128) void logits_kernel(
    const float* __restrict__ A,  // [1920,256] attn rows
    const float* __restrict__ Wf, // [256,32000]
    const float* __restrict__ bf, // [32000]
    float* __restrict__ out)      // [B,T,V]
{
    int wave = threadIdx.x >> 5, lane = threadIdx.x & 31;
    int laneM = lane & 15, kHi = lane >> 4;
    int m0 = blockIdx.y * 64;
    int n0 = blockIdx.x * 256 + wave * 64;
    v8f acc[4][4] = {};
    for (int k0 = 0; k0 < Udim; k0 += 4) {
        int kk = k0 + 2 * kHi;
        v2f a[4];
#pragma unroll
        for (int i = 0; i < 4; ++i)
            a[i] = *(const v2f*)(A + (m0 + i * 16 + laneM) * Udim + kk);
        const float* w0 = Wf + kk * Vdim;
        const float* w1 = w0 + Vdim;
#pragma unroll
        for (int j = 0; j < 4; ++j) {
            int n = n0 + j * 16 + laneM;
            v2f b; b.x = w0[n]; b.y = w1[n];
#pragma unroll
            for (int i = 0; i < 4; ++i)
                acc[i][j] = wmma_f32_4(a[i], b, acc[i][j]);
        }
    }
#pragma unroll
    for (int i = 0; i < 4; ++i)
#pragma unroll
        for (int j = 0; j < 4; ++j) {
            int n = n0 + j * 16 + laneM;
            float bias = bf[n];
#pragma unroll
            for (int r = 0; r < 8; ++r) {
                int m = m0 + i * 16 + r + 8 * kHi;
                int tt = m >> 6, bb = m & 63; // m = t*64 + b
                out[(bb * Tdec + tt) * Vdim + n] = acc[i][j][r] + bias;
            }
        }
}

// ---------------------------------------------------------------------------
extern "C" void kernel_launch(void* const* d_in, const int* in_sizes, int n_in,
                              void* d_out, int out_size, void* d_ws, size_t ws_size,
                              hipStream_t stream) {
    const int*   inputs = (const int*)d_in[0];
    const float* memory = (const float*)d_in[1];
    const float* samp_h = (const float*)d_in[2];
    const float* samp_c = (const float*)d_in[3];
    const float* emb    = (const float*)d_in[4];
    const float* Wk     = (const float*)d_in[5];
    const float* Wr     = (const float*)d_in[6];
    const float* bl     = (const float*)d_in[7];
    const float* Wm     = (const float*)d_in[8];
    const float* Wa     = (const float*)d_in[9];
    const float* Wf     = (const float*)d_in[10];
    const float* bf     = (const float*)d_in[11];
    float* out = (float*)d_out;

    // workspace layout (floats): keys | ATTN[31 slots] | h | c  == 4 MB total
    float* ws   = (float*)d_ws;
    float* keys = ws;                                  // 64*31*256
    float* ATTN = keys + Bsz * Senc * Udim;            // 31 * 64*256 (slot0=0)
    float* h    = ATTN + (Tdec + 1) * Bsz * Udim;      // 64*256
    float* c    = h + Bsz * Udim;                      // 64*256

    init_kernel<<<64, 256, 0, stream>>>(samp_h, samp_c, h, c, ATTN);
    keys_kernel<<<(Bsz * Senc) / 16, 128, 0, stream>>>(memory, Wm, keys);

    dim3 sgrid(4, 4); // (batch-group, column-group)
    for (int t = 0; t < Tdec; ++t) {
        lstm_step_kernel<<<sgrid, 256, 0, stream>>>(inputs, emb, Wk, Wr, bl,
                                                    ATTN + t * Bsz * Udim, h, c, t);
        attn_step_kernel<<<sgrid, 256, 0, stream>>>(memory, keys, Wa, h,
                                                    ATTN + (t + 1) * Bsz * Udim);
    }

    dim3 grid(Vdim / 256, (Tdec * Bsz) / 64); // 125 x 30
    logits_kernel<<<grid, 128, 0, stream>>>(ATTN + Bsz * Udim, Wf, bf, out);
}